// InvariantMessage_2473901162795
// MI455X (gfx1250) — compile-verified
//
#include <hip/hip_runtime.h>
#include <stdint.h>

typedef __attribute__((ext_vector_type(2))) float v2f;
typedef __attribute__((ext_vector_type(8))) float v8f;

#define N_EDGES  600000
#define FEAT     128
#define F3       384
#define NRBF     20
#define CUTOFF   5.0f
#define NBLOCKS  512
#define NTHREADS 256

#define W2T      132        // padded k-stride of transposed W2 (conflict-free)
#define WDT      28         // padded k-stride of transposed Wd (conflict-free)
#define SJT      132        // padded row stride of sj / h tiles (conflict-free)

// LDS layout (float offsets)
#define OFF_W2T  0                       // 384*132 = 50688
#define OFF_WDT  50688                   // 384*28  = 10752
#define OFF_B1   61440                   // 128
#define OFF_B2   61568                   // 384
#define OFF_BD   61952                   // 384
#define OFF_SJ0  62336                   // 16*132  = 2112  (gather buffer A)
#define OFF_SJ1  64448                   // 16*132  = 2112  (gather buffer B)
#define OFF_H    66560                   // 16*132  = 2112
#define OFF_RBF  68672                   // 16*20   = 320
#define LDS_FLOATS 68992                 // 275,968 bytes (~269 KB of 320 KB WGP LDS)

// Async 16B copy: global -> LDS, per-lane addresses, tracked by ASYNCcnt.
__device__ __forceinline__ void async_ld16(uint32_t lds_addr, const float* gaddr)
{
    asm volatile("global_load_async_to_lds_b128 %0, %1, off"
                 :: "v"(lds_addr), "v"(gaddr)
                 : "memory");
}

__device__ __forceinline__ void wait_async0()
{
    asm volatile("s_wait_asynccnt 0x0" ::: "memory");
}

// Load the two gather indices this thread needs for tile starting at e0.
__device__ __forceinline__ void load_idx(const long long* __restrict__ nbrs,
                                         int e0, int tid, long long jr[2])
{
    #pragma unroll
    for (int p = 0; p < 2; ++p) {
        int f = p * NTHREADS + tid;    // 0..511 float4 slots
        int r = f >> 5;                // row 0..15
        jr[p] = nbrs[2 * (long long)(e0 + r) + 1];
    }
}

// Issue this thread's two async 16B gathers into LDS buffer `buf`.
__device__ __forceinline__ void issue_gather(const float* __restrict__ s_j,
                                             const long long jr[2],
                                             float* buf, int tid)
{
    #pragma unroll
    for (int p = 0; p < 2; ++p) {
        int f  = p * NTHREADS + tid;
        int r  = f >> 5;
        int c4 = f & 31;               // float4 column
        const float* g = s_j + (long long)jr[p] * FEAT + c4 * 4;
        uint32_t l = (uint32_t)(uintptr_t)(buf + r * SJT + c4 * 4);
        async_ld16(l, g);
    }
}

__global__ __launch_bounds__(NTHREADS, 1)
void invmsg_fused_kernel(const float* __restrict__ s_j,
                         const float* __restrict__ dist,
                         const long long* __restrict__ nbrs,
                         const float* __restrict__ W1,
                         const float* __restrict__ b1,
                         const float* __restrict__ W2,
                         const float* __restrict__ b2,
                         const float* __restrict__ Wd,
                         const float* __restrict__ bd,
                         float* __restrict__ out)
{
    extern __shared__ float smem[];
    float* sW2t = smem + OFF_W2T;   // [n][k], stride W2T
    float* sWdT = smem + OFF_WDT;   // [n][k], stride WDT
    float* sb1  = smem + OFF_B1;
    float* sb2  = smem + OFF_B2;
    float* sbd  = smem + OFF_BD;
    float* sSj0 = smem + OFF_SJ0;   // [16][SJT] double-buffered gather tiles
    float* sSj1 = smem + OFF_SJ1;
    float* sH   = smem + OFF_H;     // [16][SJT]
    float* sRbf = smem + OFF_RBF;   // [16][NRBF]

    const int tid  = threadIdx.x;
    const int wave = tid >> 5;
    const int lane = tid & 31;
    const int lhi  = lane >> 4;     // 0/1 half-wave
    const int llo  = lane & 15;     // 0..15

    // ---- One-time staging: W2 and Wd transposed into LDS; biases ----
    for (int i = tid; i < (FEAT * F3) / 4; i += NTHREADS) {
        float4 w = ((const float4*)W2)[i];
        int k = (4 * i) / F3;       // 0..127
        int n = (4 * i) % F3;       // multiple of 4
        sW2t[(n + 0) * W2T + k] = w.x;
        sW2t[(n + 1) * W2T + k] = w.y;
        sW2t[(n + 2) * W2T + k] = w.z;
        sW2t[(n + 3) * W2T + k] = w.w;
    }
    for (int i = tid; i < (NRBF * F3) / 4; i += NTHREADS) {
        float4 w = ((const float4*)Wd)[i];
        int k = (4 * i) / F3;       // 0..19
        int n = (4 * i) % F3;
        sWdT[(n + 0) * WDT + k] = w.x;
        sWdT[(n + 1) * WDT + k] = w.y;
        sWdT[(n + 2) * WDT + k] = w.z;
        sWdT[(n + 3) * WDT + k] = w.w;
    }
    for (int i = tid; i < FEAT / 4; i += NTHREADS)
        ((float4*)sb1)[i] = ((const float4*)b1)[i];
    for (int i = tid; i < F3 / 4; i += NTHREADS)
        ((float4*)sb2)[i] = ((const float4*)b2)[i];
    for (int i = tid; i < F3 / 4; i += NTHREADS)
        ((float4*)sbd)[i] = ((const float4*)bd)[i];

    // ---- One-time: this wave's W1 B-fragments live in registers forever ----
    const int ncol1 = wave * 16 + llo;          // wave's GEMM1 output column
    v2f bW1[32];
    #pragma unroll
    for (int ks = 0; ks < 32; ++ks) {
        int k = 4 * ks + 2 * lhi;
        bW1[ks].x = W1[k * FEAT + ncol1];
        bW1[ks].y = W1[(k + 1) * FEAT + ncol1];
    }

    const int nIter = N_EDGES / 16;             // 37500 row tiles of 16 edges
    const int S     = (int)gridDim.x;

    // ---- Prologue: kick off async gather of first tile; pre-load next indices ----
    long long jr[2];                            // indices for tile it+S (one ahead)
    int it = blockIdx.x;
    if (it < nIter) {
        load_idx(nbrs, it * 16, tid, jr);
        issue_gather(s_j, jr, sSj0, tid);       // uses jr for tile `it`
        if (it + S < nIter) load_idx(nbrs, (it + S) * 16, tid, jr);
    }
    __syncthreads();                            // weights staged

    int cur = 0;
    for (; it < nIter; it += S, cur ^= 1) {
        const int e0 = it * 16;
        float* sjc = cur ? sSj1 : sSj0;         // current gather buffer
        float* sjn = cur ? sSj0 : sSj1;         // next gather buffer

        // ---- our async gathers for this tile have landed; make LDS visible ----
        wait_async0();
        __syncthreads();

        // ---- issue next tile's gather into the other buffer (overlaps GEMMs) ----
        int itn = it + S;
        if (itn < nIter) {
            issue_gather(s_j, jr, sjn, tid);    // jr holds indices for tile itn
            if (itn + S < nIter) load_idx(nbrs, (itn + S) * 16, tid, jr);
        }

        // ---- rbf table for this tile (published by the phase-2 barrier) ----
        for (int q = tid; q < 16 * NRBF; q += NTHREADS) {
            int r = q / NRBF, i = q % NRBF;
            float d   = dist[e0 + r];
            float arg = (float)(i + 1) * (3.14159265358979323846f / CUTOFF) * d;
            sRbf[r * NRBF + i] = __sinf(arg) / d;
        }

        // ---- Phase 2: h = SiLU(sj @ W1 + b1); B fragments from registers ----
        {
            v8f acc = {};
            #pragma unroll
            for (int ks = 0; ks < 32; ++ks) {
                v2f a = *(const v2f*)(sjc + llo * SJT + 4 * ks + 2 * lhi);
                acc = __builtin_amdgcn_wmma_f32_16x16x4_f32(
                    false, a, false, bW1[ks], (short)0, acc, false, false);
            }
            float bn = sb1[ncol1];
            #pragma unroll
            for (int r = 0; r < 8; ++r) {
                float x = acc[r] + bn;
                float h = x / (1.0f + __expf(-x));   // x * sigmoid(x)
                sH[(r + 8 * lhi) * SJT + ncol1] = h;
            }
        }
        __syncthreads();

        // ---- Phase 3: phi = h @ W2 + b2 ; ws = rbf @ Wd + bd ; out = phi*ws ----
        {
            v2f aR[5];
            #pragma unroll
            for (int ks = 0; ks < 5; ++ks)
                aR[ks] = *(const v2f*)(sRbf + llo * NRBF + 4 * ks + 2 * lhi);

            #pragma unroll
            for (int t = 0; t < 3; ++t) {
                const int ncol = (wave + 8 * t) * 16 + llo;

                v8f acc = {};
                #pragma unroll
                for (int ks = 0; ks < 32; ++ks) {
                    v2f a = *(const v2f*)(sH + llo * SJT + 4 * ks + 2 * lhi);
                    v2f b = *(const v2f*)(sW2t + ncol * W2T + 4 * ks + 2 * lhi);
                    acc = __builtin_amdgcn_wmma_f32_16x16x4_f32(
                        false, a, false, b, (short)0, acc, false, false);
                }

                v8f accw = {};
                #pragma unroll
                for (int ks = 0; ks < 5; ++ks) {
                    v2f b = *(const v2f*)(sWdT + ncol * WDT + 4 * ks + 2 * lhi);
                    accw = __builtin_amdgcn_wmma_f32_16x16x4_f32(
                        false, aR[ks], false, b, (short)0, accw, false, false);
                }

                float b2n = sb2[ncol];
                float bdn = sbd[ncol];
                #pragma unroll
                for (int r = 0; r < 8; ++r) {
                    int m = r + 8 * lhi;
                    float phi = acc[r] + b2n;
                    float ws  = accw[r] + bdn;
                    out[(long long)(e0 + m) * F3 + ncol] = phi * ws;
                }
            }
        }
        __syncthreads();
    }
}

extern "C" void kernel_launch(void* const* d_in, const int* in_sizes, int n_in,
                              void* d_out, int out_size, void* d_ws, size_t ws_size,
                              hipStream_t stream)
{
    const float*     s_j  = (const float*)d_in[0];
    const float*     dist = (const float*)d_in[1];
    const long long* nbrs = (const long long*)d_in[2];
    const float*     W1   = (const float*)d_in[3];
    const float*     b1   = (const float*)d_in[4];
    const float*     W2   = (const float*)d_in[5];
    const float*     b2   = (const float*)d_in[6];
    const float*     Wd   = (const float*)d_in[7];
    const float*     bd   = (const float*)d_in[8];
    float*           out  = (float*)d_out;

    size_t shmem = (size_t)LDS_FLOATS * sizeof(float);
    invmsg_fused_kernel<<<NBLOCKS, NTHREADS, shmem, stream>>>(
        s_j, dist, nbrs, W1, b1, W2, b2, Wd, bd, out);
}